// PauliGate_1597727834808
// MI455X (gfx1250) — compile-verified
//
#include <hip/hip_runtime.h>

typedef float v2f __attribute__((ext_vector_type(2)));
typedef float v8f __attribute__((ext_vector_type(8)));

// Apply 2x2 pauli P to qubit 0 of a (2,)*23 + (8,) f32 state.
// Flat layout: x = state[0 .. HALF), y = state[HALF .. 2*HALF), HALF = 2^25.
//   out0[k] = P00*x[k] + P01*y[k]
//   out1[k] = P10*x[k] + P11*y[k]
//
// OUT(2 x HALF) = P(2x2) * S(2 x HALF) via V_WMMA_F32_16X16X4_F32, with the
// Pauli embedded in A TWICE to make all 32 lanes carry unique data:
//   A[0..1, 0..1] = P   (lanes 0-15 hold A cols K=0,1; D rows 0,1 -> lanes 0-15)
//   A[8..9, 2..3] = P   (lanes 16-31 hold A cols K=2,3; D rows 8,9 -> lanes 16-31)
// B per 32-column span: lane L holds b.x = x[base+L] (rows K=0 / K=2) and
// b.y = y[base+L] (rows K=1 / K=3) -- fully contiguous full-wave loads.
// Result: D VGPR0 lane L = out0[base+L], D VGPR1 lane L = out1[base+L] for
// ALL 32 lanes -> fully dense contiguous stores, no EXEC predication.
//
// Memory-bound: 512 MB total traffic -> ~22 us floor @ 23.3 TB/s. Every byte
// is touched exactly once, so all loads/stores are non-temporal to keep the
// 192 MB L2 from being churned.
__global__ void pauli_q0_wmma_kernel(const float* __restrict__ state,
                                     const float* __restrict__ pauli,
                                     float* __restrict__ out,
                                     int half, int nSpans) {
  const int lane   = threadIdx.x & 31;
  const int wave   = (int)((blockIdx.x * blockDim.x + threadIdx.x) >> 5);
  const int nWaves = (int)((gridDim.x * blockDim.x) >> 5);

  // --- Build A (wave-invariant) -------------------------------------------
  // lane 0:  A[0,0]=p00, A[0,1]=p01      lane 24: A[8,2]=p00, A[8,3]=p01
  // lane 1:  A[1,0]=p10, A[1,1]=p11      lane 25: A[9,2]=p10, A[9,3]=p11
  // all other lanes: 0
  const float p00 = pauli[0], p01 = pauli[1];
  const float p10 = pauli[2], p11 = pauli[3];
  const bool row0 = (lane == 0) || (lane == 24);
  const bool row1 = (lane == 1) || (lane == 25);
  v2f a;
  a.x = row0 ? p00 : (row1 ? p10 : 0.0f);
  a.y = row0 ? p01 : (row1 ? p11 : 0.0f);

  const float* __restrict__ xh = state;         // j = 0 half
  const float* __restrict__ yh = state + half;  // j = 1 half
  float* __restrict__ o0 = out;
  float* __restrict__ o1 = out + half;

  // One WMMA covers a span of 32 consecutive columns. Unroll 4 spans per
  // iteration: batch 8 NT loads (512 B from x, 512 B from y), then 4 WMMAs,
  // then 8 NT stores -- maximizes loads in flight per wave.
  for (int s = wave * 4; s < nSpans; s += nWaves * 4) {
    const int base = s * 32 + lane;

    v2f b[4];
#pragma unroll
    for (int u = 0; u < 4; ++u) {
      b[u].x = __builtin_nontemporal_load(xh + base + u * 32);
      b[u].y = __builtin_nontemporal_load(yh + base + u * 32);
    }

    v8f d[4];
#pragma unroll
    for (int u = 0; u < 4; ++u) {
      v8f c = {};
      // EXEC is all-1s here: all control flow is wave-uniform.
      d[u] = __builtin_amdgcn_wmma_f32_16x16x4_f32(
          /*neg_a=*/false, a, /*neg_b=*/false, b[u],
          /*c_mod=*/(short)0, c, /*reuse_a=*/false, /*reuse_b=*/false);
    }

#pragma unroll
    for (int u = 0; u < 4; ++u) {
      __builtin_nontemporal_store(d[u][0], o0 + base + u * 32);  // out0 row
      __builtin_nontemporal_store(d[u][1], o1 + base + u * 32);  // out1 row
    }
  }
}

extern "C" void kernel_launch(void* const* d_in, const int* in_sizes, int n_in,
                              void* d_out, int out_size, void* d_ws, size_t ws_size,
                              hipStream_t stream) {
  (void)n_in; (void)out_size; (void)d_ws; (void)ws_size;
  const float* state = (const float*)d_in[0];
  const float* pauli = (const float*)d_in[1];
  float* out = (float*)d_out;

  const int half   = in_sizes[0] / 2;  // 2^25 elements per qubit-0 half
  const int nSpans = half / 32;        // 32-column spans per WMMA (= 2^20)

  // 2048 blocks x 256 threads = 16384 wave32s; 4 spans/wave/iter ->
  // 16 grid-stride iterations; 8 NT loads in flight per wave per iteration.
  dim3 grid(2048), block(256);
  pauli_q0_wmma_kernel<<<grid, block, 0, stream>>>(state, pauli, out,
                                                   half, nSpans);
}